// DiffusionBlock_32418413150438
// MI455X (gfx1250) — compile-verified
//
#include <hip/hip_runtime.h>

#ifndef __has_builtin
#define __has_builtin(x) 0
#endif

#define AS1 __attribute__((address_space(1)))
#define AS3 __attribute__((address_space(3)))

// ---- problem constants (from reference) ----
#define BN 16
#define CN 3
#define HN 512
#define WN 512
#define HP 514           // H+2
#define WP 514
#define TX 32            // output tile width
#define TY 8             // output tile height
#define VX (TX + 1)      // v-grid tile width  (33)
#define VY (TY + 1)      // v-grid tile height (9)
#define UX (TX + 2)      // u-halo tile width  (34)
#define UY (TY + 2)      // u-halo tile height (10)
#define NTHREADS 256

// ---- CDNA5 async global->LDS copy (per-lane gather into LDS) ----
__device__ __forceinline__ void async_copy_b32(const float* g, float* l) {
#if __has_builtin(__builtin_amdgcn_global_load_async_to_lds_b32)
  __builtin_amdgcn_global_load_async_to_lds_b32((AS1 int*)g, (AS3 int*)l, 0, 0);
#else
  unsigned loff = (unsigned)(unsigned long long)(AS3 void*)l;
  unsigned long long ga = (unsigned long long)g;
  asm volatile("global_load_async_to_lds_b32 %0, %1, off"
               :: "v"(loff), "v"(ga) : "memory");
#endif
}

__device__ __forceinline__ void wait_asynccnt0() {
#if __has_builtin(__builtin_amdgcn_s_wait_asynccnt)
  __builtin_amdgcn_s_wait_asynccnt(0);
#else
  asm volatile("s_wait_asynccnt 0" ::: "memory");
#endif
}

__global__ __launch_bounds__(NTHREADS)
void diffusion_fused_kernel(const float* __restrict__ u,
                            const float* __restrict__ a,
                            const float* __restrict__ b,
                            const float* __restrict__ c,
                            const float* __restrict__ alpha,
                            const float* __restrict__ g1,
                            const float* __restrict__ g2,
                            const float* __restrict__ g3,
                            const float* __restrict__ g4,
                            const float* __restrict__ taup,
                            float* __restrict__ out) {
  __shared__ float su[UY][UX];        // clamped u halo tile
  __shared__ float sv[4][VY][VX];     // v1..v4 intermediate fields
  __shared__ float sk[16];            // taps: g1[0..3] g2[0..3] g3[0..3] g4[0..3]
  __shared__ float stau;

  const int t     = threadIdx.x;                 // 0..255
  const int x0    = blockIdx.x * TX;
  const int y0    = blockIdx.y * TY;
  const int plane = blockIdx.z;                  // 0..B*C-1
  const int ch    = plane % CN;

  const float* uplane = u + (size_t)plane * (HN * WN);
  const size_t wbase  = (size_t)plane * (HP * WP);

  // ---- phase 0: taps + tau into LDS ----
  if (t < 4)        sk[t] = g1[ch * 4 + t];
  else if (t < 8)   sk[t] = g2[ch * 4 + (t - 4)];
  else if (t < 12)  sk[t] = g3[ch * 4 + (t - 8)];
  else if (t < 16)  sk[t] = g4[ch * 4 + (t - 12)];
  if (t == 16)      stau  = taup[0];

  // ---- phase 0b: async-gather u halo tile (replication-pad clamp) into LDS ----
  for (int idx = t; idx < UY * UX; idx += NTHREADS) {
    int r  = idx / UX;
    int cc = idx % UX;
    int gy = y0 - 1 + r;  gy = gy < 0 ? 0 : (gy > HN - 1 ? HN - 1 : gy);
    int gx = x0 - 1 + cc; gx = gx < 0 ? 0 : (gx > WN - 1 ? WN - 1 : gx);
    async_copy_b32(uplane + (size_t)gy * WN + gx, &su[0][0] + idx);
  }
  wait_asynccnt0();
  __syncthreads();

  const float k100 = sk[0],  k101 = sk[1],  k110 = sk[2],  k111 = sk[3];
  const float k200 = sk[4],  k201 = sk[5],  k210 = sk[6],  k211 = sk[7];
  const float k300 = sk[8],  k301 = sk[9],  k310 = sk[10], k311 = sk[11];
  const float k400 = sk[12], k401 = sk[13], k410 = sk[14], k411 = sk[15];
  const float tau  = stau;

  // ---- phase A: first convs + reweighting -> v1..v4 on the (VY x VX) grid ----
  for (int idx = t; idx < VY * VX; idx += NTHREADS) {
    int il = idx / VX;
    int jl = idx % VX;

    float u00 = su[il][jl],     u01 = su[il][jl + 1];
    float u10 = su[il + 1][jl], u11 = su[il + 1][jl + 1];

    float ux1 = fmaf(k100, u00, fmaf(k101, u01, fmaf(k110, u10, k111 * u11)));
    float ux2 = fmaf(k200, u00, fmaf(k201, u01, fmaf(k210, u10, k211 * u11)));
    float uy1 = fmaf(k300, u00, fmaf(k301, u01, fmaf(k310, u10, k311 * u11)));
    float uy2 = fmaf(k400, u00, fmaf(k401, u01, fmaf(k410, u10, k411 * u11)));

    size_t wi = wbase + (size_t)(y0 + il) * WP + (size_t)(x0 + jl);
    float Av = a[wi], Bv = b[wi], Cv = c[wi], al = alpha[wi];

    float s    = (Bv > 0.0f) ? 1.0f : ((Bv < 0.0f) ? -1.0f : 0.0f);
    float beta = (1.0f - 2.0f * al) * s;
    float w1a  = Av * (1.0f - al)   * 0.5f;
    float w2a  = Av * al            * 0.5f;
    float w1b  = Bv * (1.0f - beta) * 0.25f;
    float w2b  = Bv * (1.0f + beta) * 0.25f;
    float w1c  = Cv * (1.0f - al)   * 0.5f;
    float w2c  = Cv * al            * 0.5f;

    sv[0][il][jl] = fmaf(w1a, ux1, fmaf(w2a, ux2, fmaf(w1b, uy1, w2b * uy2)));
    sv[1][il][jl] = fmaf(w2a, ux1, fmaf(w1a, ux2, fmaf(w2b, uy1, w1b * uy2)));
    sv[2][il][jl] = fmaf(w1b, ux1, fmaf(w2b, ux2, fmaf(w1c, uy1, w2c * uy2)));
    sv[3][il][jl] = fmaf(w2b, ux1, fmaf(w1b, ux2, fmaf(w2c, uy1, w1c * uy2)));
  }
  __syncthreads();

  // ---- phase B: second convs with neg_flip(k): tap at (di,dj) is -k[1-di][1-dj] ----
  int tx = t % TX;
  int ty = t / TX;

  float acc;
  acc  = fmaf(sv[0][ty][tx],     k111, fmaf(sv[0][ty][tx + 1],     k110,
         fmaf(sv[0][ty + 1][tx], k101,      sv[0][ty + 1][tx + 1] * k100)));
  acc  = fmaf(sv[1][ty][tx],     k211, fmaf(sv[1][ty][tx + 1],     k210,
         fmaf(sv[1][ty + 1][tx], k201, fmaf(sv[1][ty + 1][tx + 1], k200, acc))));
  acc  = fmaf(sv[2][ty][tx],     k311, fmaf(sv[2][ty][tx + 1],     k310,
         fmaf(sv[2][ty + 1][tx], k301, fmaf(sv[2][ty + 1][tx + 1], k300, acc))));
  acc  = fmaf(sv[3][ty][tx],     k411, fmaf(sv[3][ty][tx + 1],     k410,
         fmaf(sv[3][ty + 1][tx], k401, fmaf(sv[3][ty + 1][tx + 1], k400, acc))));

  float uc = su[ty + 1][tx + 1];                      // u(y0+ty, x0+tx)
  float res = fmaf(-tau, acc, uc);                    // u + tau * (-(sum)) = u + tau*Au

  out[(size_t)plane * (HN * WN) + (size_t)(y0 + ty) * WN + (size_t)(x0 + tx)] = res;
}

extern "C" void kernel_launch(void* const* d_in, const int* in_sizes, int n_in,
                              void* d_out, int out_size, void* d_ws, size_t ws_size,
                              hipStream_t stream) {
  const float* u     = (const float*)d_in[0];
  const float* a     = (const float*)d_in[1];
  const float* b     = (const float*)d_in[2];
  const float* c     = (const float*)d_in[3];
  const float* alpha = (const float*)d_in[4];
  const float* g1    = (const float*)d_in[5];
  const float* g2    = (const float*)d_in[6];
  const float* g3    = (const float*)d_in[7];
  const float* g4    = (const float*)d_in[8];
  const float* tau   = (const float*)d_in[9];
  float* out         = (float*)d_out;

  dim3 grid(WN / TX, HN / TY, BN * CN);   // 16 x 64 x 48 blocks
  dim3 block(NTHREADS);
  diffusion_fused_kernel<<<grid, block, 0, stream>>>(u, a, b, c, alpha,
                                                     g1, g2, g3, g4, tau, out);
}